// BKTModel_41214506173092
// MI455X (gfx1250) — compile-verified
//
#include <hip/hip_runtime.h>
#include <math.h>

#define N_KCS 100
#define N_OBS 1000
#define N_AB  30
#define BATCH 512
#define TLEN  100

typedef float v2f __attribute__((ext_vector_type(2)));
typedef float v8f __attribute__((ext_vector_type(8)));

__device__ __forceinline__ float sigf(float x) {
    return 1.0f / (1.0f + __expf(-x));
}

__device__ __forceinline__ float wave_max32(float v) {
#pragma unroll
    for (int o = 16; o >= 1; o >>= 1) v = fmaxf(v, __shfl_xor(v, o, 32));
    return v;
}
__device__ __forceinline__ float wave_sum32(float v) {
#pragma unroll
    for (int o = 16; o >= 1; o >>= 1) v += __shfl_xor(v, o, 32);
    return v;
}

// ---------------------------------------------------------------------------
// Kernel 1: D = A (1000x100, one-hot rows) @ Bmat (100x16) via
// v_wmma_f32_16x16x4_f32.  Bmat columns 0..4 = kc_logits columns, column 5 =
// iota(k).  Since A rows are one-hot, D[o,0..4] = kc_logits[kc(o)] and
// D[o,5] = (float)kc(o).  Result stored as obs_params[o*8 + col].
// One wave (32 threads) per 16-row tile of D.  B-fragment construction is
// branch-free (clamped load + v_cndmask selects) so EXEC stays all-1s and no
// saveexec churn surrounds the WMMA chain.
// ---------------------------------------------------------------------------
__global__ __launch_bounds__(32) void bkt_obs_params_wmma(
    const float* __restrict__ A, const float* __restrict__ kc_logits,
    float* __restrict__ obs_params)
{
    const int lane = threadIdx.x;
    const int half = lane >> 4;      // 0: this lane holds K=k0+0,1   1: K=k0+2,3
    const int l15  = lane & 15;
    const int row  = blockIdx.x * 16 + l15;                 // A-fragment row (M)
    const int rcl  = row < N_OBS ? row : (N_OBS - 1);       // clamped (in-bounds)
    const float rm = row < N_OBS ? 1.0f : 0.0f;
    const int ncl  = l15 < 5 ? l15 : 4;                     // clamped B column

    v8f acc = {};
#pragma unroll
    for (int k0 = 0; k0 < N_KCS; k0 += 4) {
        const int ka = k0 + half * 2;
        // A 16x4 fragment, 8B-aligned vector load (ka even):
        // lane L<16 holds A[M=L, k0+0..1]; lane L>=16 holds A[M=L-16, k0+2..3].
        v2f av = *(const v2f*)(A + rcl * N_KCS + ka);
        av.x *= rm;
        av.y *= rm;
        // B 4x16 fragment: B[k][n] = (n<5) ? kc_logits[k][n] : (n==5) ? k : 0.
        // Unconditional clamped loads + cndmask selects (no divergence).
        const float r0 = kc_logits[(ka + 0) * 5 + ncl];
        const float r1 = kc_logits[(ka + 1) * 5 + ncl];
        v2f bv;
        bv.x = (l15 < 5) ? r0 : ((l15 == 5) ? (float)(ka + 0) : 0.0f);
        bv.y = (l15 < 5) ? r1 : ((l15 == 5) ? (float)(ka + 1) : 0.0f);

        acc = __builtin_amdgcn_wmma_f32_16x16x4_f32(
                false, av, false, bv, (short)0, acc, false, false);
    }

    // D layout: VGPR j -> (M=j, N=lane) for lanes 0-15, (M=j+8, N=lane-16) hi.
#pragma unroll
    for (int j = 0; j < 8; ++j) {
        const int orow = blockIdx.x * 16 + j + half * 8;
        if (l15 < 6 && orow < N_OBS) obs_params[orow * 8 + l15] = acc[j];
    }
}

// ---------------------------------------------------------------------------
// Kernel 2: the sequential BKT scan.  One wave32 per batch element; lanes =
// ability grid points (30 active, 2 inert).  Per-KC state (100 x 30) lives in
// LDS — the one-hot structure of A collapses the reference's (B,1000,30)
// state tensor and the 1000x1000 M matrix into per-KC rows (1000x less
// per-step update work, 61 MB -> 12.8 KB state per student).
// ---------------------------------------------------------------------------
__global__ __launch_bounds__(32) void bkt_scan(
    const int* __restrict__ prev_kc, const int* __restrict__ curr_kc,
    const float* __restrict__ prev_corr, const float* __restrict__ kc_logits,
    const float* __restrict__ comp_w, const float* __restrict__ comp_mu,
    const float* __restrict__ comp_lv, const float* __restrict__ obs_params,
    float* __restrict__ out)
{
    __shared__ float state[N_KCS][32];   // [kc][lane], 12.8 KB
    const int lane = threadIdx.x;
    const int b = blockIdx.x;
    const float ab = -3.0f + (float)lane * (6.0f / 29.0f);   // linspace(-3,3,30)

    // ---- GMM log-prior over abilities (replicates reference gmm_logpdf) ----
    float w[5], mu[5], lv[5];
#pragma unroll
    for (int j = 0; j < 5; ++j) { w[j] = comp_w[j]; mu[j] = comp_mu[j]; lv[j] = comp_lv[j]; }
    float wm = w[0];
#pragma unroll
    for (int j = 1; j < 5; ++j) wm = fmaxf(wm, w[j]);
    float wsum = 0.0f;
#pragma unroll
    for (int j = 0; j < 5; ++j) wsum += __expf(w[j] - wm);
    const float lsw = wm + __logf(wsum);   // logsumexp(comp_w)

    float tv[5];
#pragma unroll
    for (int j = 0; j < 5; ++j) {
        const float dv = __expf(lv[j]);
        const float d  = ab - mu[j];
        // NOTE: reference uses +0.5*d^2/var (sic) - log(sqrt(2*pi*var))
        tv[j] = 0.5f * d * d / dv - __logf(sqrtf(6.28318530717958647692f * dv))
                + (w[j] - lsw);
    }
    float tm = tv[0];
#pragma unroll
    for (int j = 1; j < 5; ++j) tm = fmaxf(tm, tv[j]);
    float tsum = 0.0f;
#pragma unroll
    for (int j = 0; j < 5; ++j) tsum += __expf(tv[j] - tm);
    float ability = tm + __logf(tsum);
    if (lane >= N_AB) ability = -1e30f;    // inert lanes: softmax weight -> 0

    // ---- state init: sigmoid(kc_logits[k,4]) broadcast over ability axis ----
    for (int k = 0; k < N_KCS; ++k) state[k][lane] = sigf(kc_logits[k * 5 + 4]);

    // ---- t = 0 ----
    const int ck0 = curr_kc[b * TLEN];
    const float cl2_0 = obs_params[ck0 * 8 + 2];
    const float cl3_0 = obs_params[ck0 * 8 + 3];
    const int   kcc0  = (int)(obs_params[ck0 * 8 + 5] + 0.5f);
    const float cs0   = state[kcc0][lane];
    float pca = sigf(cl2_0 + ab) * (1.0f - cs0) + sigf(cl3_0 + ab) * cs0;
    {
        const float m   = wave_max32(ability);
        const float e   = __expf(ability - m);
        const float den = wave_sum32(e);
        const float num = wave_sum32(e * pca);
        if (lane == 0) out[b * TLEN] = num / den;
    }

    // ---- sequential scan over t = 1..T-1 ----
    for (int t = 1; t < TLEN; ++t) {
        const int   pk = prev_kc[b * TLEN + t];
        const int   ck = curr_kc[b * TLEN + t];
        const float c  = prev_corr[b * TLEN + t];   // exactly 0.0 or 1.0

        ability += c * __logf(pca) + (1.0f - c) * __logf(1.0f - pca);

        const float pl0 = obs_params[pk * 8 + 0];
        const float pl1 = obs_params[pk * 8 + 1];
        const float pl2 = obs_params[pk * 8 + 2];
        const float pl3 = obs_params[pk * 8 + 3];
        const int   kcp = (int)(obs_params[pk * 8 + 5] + 0.5f);

        const float p0  = sigf(pl2 + ab);
        const float p1  = sigf(pl3 + ab);
        // p^c * (1-p)^(1-c) with c in {0,1} == linear select (exact)
        const float po0 = c * p0 + (1.0f - c) * (1.0f - p0);
        const float po1 = c * p1 + (1.0f - c) * (1.0f - p1);
        const float s   = state[kcp][lane];
        const float filt = po1 * s / (po0 * (1.0f - s) + po1 * s);
        const float plearn  = sigf(pl0);
        const float pforget = sigf(pl1);
        const float pred = plearn * (1.0f - filt) + (1.0f - pforget) * filt;
        state[kcp][lane] = pred;    // M[pk]-masked update collapses to one row

        const float cl2 = obs_params[ck * 8 + 2];
        const float cl3 = obs_params[ck * 8 + 3];
        const int   kcc = (int)(obs_params[ck * 8 + 5] + 0.5f);
        const float cs  = state[kcc][lane];   // same-wave LDS ops are in-order
        pca = sigf(cl2 + ab) * (1.0f - cs) + sigf(cl3 + ab) * cs;

        const float m   = wave_max32(ability);
        const float e   = __expf(ability - m);
        const float den = wave_sum32(e);
        const float num = wave_sum32(e * pca);
        if (lane == 0) out[b * TLEN + t] = num / den;
    }
}

extern "C" void kernel_launch(void* const* d_in, const int* in_sizes, int n_in,
                              void* d_out, int out_size, void* d_ws, size_t ws_size,
                              hipStream_t stream) {
    const int*   prev_kc   = (const int*)  d_in[0];
    const int*   curr_kc   = (const int*)  d_in[1];
    const float* prev_corr = (const float*)d_in[2];
    const float* A         = (const float*)d_in[3];
    const float* kc_logits = (const float*)d_in[4];
    const float* comp_w    = (const float*)d_in[5];
    const float* comp_mu   = (const float*)d_in[6];
    const float* comp_lv   = (const float*)d_in[7];
    float* out        = (float*)d_out;
    float* obs_params = (float*)d_ws;      // N_OBS * 8 floats = 32 KB scratch

    // Precompute per-obs params with WMMA f32 (63 tiles of 16 rows).
    bkt_obs_params_wmma<<<dim3((N_OBS + 15) / 16), dim3(32), 0, stream>>>(
        A, kc_logits, obs_params);

    // One wave32 per student.
    bkt_scan<<<dim3(BATCH), dim3(32), 0, stream>>>(
        prev_kc, curr_kc, prev_corr, kc_logits, comp_w, comp_mu, comp_lv,
        obs_params, out);
}